// AttentionLayer_43447889166944
// MI455X (gfx1250) — compile-verified
//
#include <hip/hip_runtime.h>
#include <math.h>

#define B_   64
#define T_   2000
#define DQ_  1024
#define DM_  512
#define D_   128
#define F_   32
#define KK_  31
#define WIN_ 7
#define NEGV (-4294967295.0f)   // -2^32 + 1

typedef float v2f __attribute__((ext_vector_type(2)));
typedef float v8f __attribute__((ext_vector_type(8)));

// ---------------------------------------------------------------------------
// Kernel 1: pack Wm[512,128] into V_WMMA_F32_16X16X4_F32 B-fragment order.
// Layout: pB[((kstep*8 + ntile)*32 + lane)*2 + v] = Wm[(kstep*4 + v + 2*(lane>=16))*128
//                                                      + ntile*16 + (lane&15)]
// so the GEMM inner loop reads one contiguous float2 per lane (256B per wave).
// ---------------------------------------------------------------------------
__global__ void __launch_bounds__(256) pack_wm(const float* __restrict__ Wm,
                                               float* __restrict__ pB) {
    int tid = blockIdx.x * blockDim.x + threadIdx.x;   // 0 .. 128*8*32-1
    if (tid >= 128 * 8 * 32) return;
    int lane  = tid & 31;
    int nt    = (tid >> 5) & 7;
    int kstep = tid >> 8;
    int koff  = (lane >> 4) << 1;
    int n     = nt * 16 + (lane & 15);
    int k0    = kstep * 4 + koff;
    pB[tid * 2 + 0] = Wm[(k0 + 0) * D_ + n];
    pB[tid * 2 + 1] = Wm[(k0 + 1) * D_ + n];
}

// ---------------------------------------------------------------------------
// Kernel 2: values[B*T,128] = memory[B*T,512] @ Wm[512,128] via fp32 WMMA.
// One wave per 16-row block; 8 accumulator tiles cover the full N=128 so each
// row of `memory` is streamed from HBM exactly once (HBM-bound: ~330MB total).
// ---------------------------------------------------------------------------
__global__ void __launch_bounds__(256) values_gemm(const float* __restrict__ mem,
                                                   const float* __restrict__ pB,
                                                   float* __restrict__ values) {
    const int lane = threadIdx.x & 31;
    const int wave = blockIdx.x * (blockDim.x >> 5) + (threadIdx.x >> 5);
    const int row0 = wave * 16;
    if (row0 >= B_ * T_) return;

    // A fragment addressing (16x4 f32): lane&15 -> M, lane>=16 -> K+=2
    const int m    = lane & 15;
    const int koff = (lane >> 4) << 1;
    const float* arow = mem + (size_t)(row0 + m) * DM_ + koff;

    v8f c[8];
#pragma unroll
    for (int i = 0; i < 8; ++i) c[i] = (v8f)(0.0f);

    for (int kk = 0; kk < DM_ / 4; ++kk) {
        v2f a = *(const v2f*)(arow + kk * 4);
        if ((kk & 15) == 0)
            __builtin_prefetch(arow + kk * 4 + 64, 0, 1);  // global_prefetch_b8
        const float* bbase = pB + (size_t)kk * 8 * 64 + lane * 2;
#pragma unroll
        for (int nt = 0; nt < 8; ++nt) {
            v2f bf = *(const v2f*)(bbase + nt * 64);
            c[nt] = __builtin_amdgcn_wmma_f32_16x16x4_f32(
                false, a, false, bf, (short)0, c[nt], false, false);
        }
    }

    // C/D layout: VGPR r, lanes 0-15 -> M=r, lanes 16-31 -> M=r+8; N = lane&15
    const int rsel = (lane >> 4) * 8;
    const int col  = lane & 15;
#pragma unroll
    for (int nt = 0; nt < 8; ++nt) {
#pragma unroll
        for (int r = 0; r < 8; ++r) {
            values[(size_t)(row0 + r + rsel) * D_ + nt * 16 + col] = c[nt][r];
        }
    }
}

// ---------------------------------------------------------------------------
// Kernel 3: pq[b,d] = query[b,:] @ Wq[:,d]   (tiny: 17 MFLOP)
// ---------------------------------------------------------------------------
__global__ void __launch_bounds__(128) pq_kernel(const float* __restrict__ query,
                                                 const float* __restrict__ Wq,
                                                 float* __restrict__ pq) {
    __shared__ float qs[DQ_];
    const int b = blockIdx.x, d = threadIdx.x;
    for (int i = d; i < DQ_; i += 128) qs[i] = query[b * DQ_ + i];
    __syncthreads();
    float acc = 0.f;
    for (int q = 0; q < DQ_; ++q) acc = fmaf(qs[q], Wq[q * D_ + d], acc);
    pq[b * D_ + d] = acc;
}

// ---------------------------------------------------------------------------
// Kernel 4: conv1d(SAME) features -> ploc -> additive tanh energy.
// One block = one b, 16 consecutive t; Wloc/conv_kernel staged in LDS;
// reduction over D via wave32 shfl_xor + 4-partial LDS combine.
// ---------------------------------------------------------------------------
__global__ void __launch_bounds__(128) energy_kernel(
    const float* __restrict__ prev_align, const float* __restrict__ ck,
    const float* __restrict__ cb, const float* __restrict__ Wloc,
    const float* __restrict__ sv, const float* __restrict__ sb,
    const float* __restrict__ pq, const float* __restrict__ values,
    float* __restrict__ energy) {
    __shared__ float WlocS[F_ * D_];
    __shared__ float sf[16][F_];
    __shared__ float ckS[KK_ * F_];
    __shared__ float winS[46];
    __shared__ float pqS[D_], svS[D_], sbS[D_], cbS[F_];
    __shared__ float part[4];

    const int tid = threadIdx.x;
    const int b   = blockIdx.x / (T_ / 16);
    const int t0  = (blockIdx.x % (T_ / 16)) * 16;

    for (int i = tid; i < F_ * D_; i += 128) WlocS[i] = Wloc[i];
    for (int i = tid; i < KK_ * F_; i += 128) ckS[i] = ck[i];
    if (tid < F_) cbS[tid] = cb[tid];
    pqS[tid] = pq[b * D_ + tid];
    svS[tid] = sv[tid];
    sbS[tid] = sb[tid];
    for (int i = tid; i < 46; i += 128) {
        int g = t0 - 15 + i;
        winS[i] = (g >= 0 && g < T_) ? prev_align[b * T_ + g] : 0.f;
    }
    __syncthreads();

    // conv features for 16 t's: f[tt][ff] = bias + sum_tap win[tt+tap]*k[tap][ff]
    for (int task = tid; task < 16 * F_; task += 128) {
        int tt = task >> 5, ff = task & 31;
        float s = cbS[ff];
#pragma unroll
        for (int tap = 0; tap < KK_; ++tap)
            s = fmaf(winS[tt + tap], ckS[tap * F_ + ff], s);
        sf[tt][ff] = s;
    }
    __syncthreads();

    const int lane = tid & 31, wid = tid >> 5;
    for (int tt = 0; tt < 16; ++tt) {
        float pl = 0.f;
#pragma unroll
        for (int ff = 0; ff < F_; ++ff)
            pl = fmaf(sf[tt][ff], WlocS[ff * D_ + tid], pl);
        float v = values[(size_t)(b * T_ + t0 + tt) * D_ + tid];
        float e = svS[tid] * tanhf(v + pqS[tid] + pl + sbS[tid]);
        for (int off = 16; off > 0; off >>= 1) e += __shfl_xor(e, off, 32);
        if (lane == 0) part[wid] = e;
        __syncthreads();
        if (tid == 0) energy[b * T_ + t0 + tt] = part[0] + part[1] + part[2] + part[3];
        __syncthreads();
    }
}

// ---------------------------------------------------------------------------
// Kernel 5: monotonic-window + length mask, softmax, argmax (first-index ties),
// attention matvec, cumulative next_state. One block per b.
// Allowed t:  t >= pma  &&  t <= pma+WIN-1 (via reverse mask)  &&  t < memlen
// ---------------------------------------------------------------------------
__global__ void __launch_bounds__(256) softmax_attn(
    const float* __restrict__ energy, const float* __restrict__ prev_align,
    const int* __restrict__ pma, const int* __restrict__ memlen,
    const float* __restrict__ values, float* __restrict__ out_att,
    float* __restrict__ out_align, float* __restrict__ out_next,
    float* __restrict__ out_maxatt) {
    __shared__ float alignS[T_];
    __shared__ float red[256];
    __shared__ int   redi[256];

    const int b = blockIdx.x, tid = threadIdx.x;
    const int p  = pma[b];
    const int ml = memlen[b];
    const int mask_len = max(0, T_ - WIN_ - p);

    // pass 1: masked energy + max/argmax
    float mx = NEGV; int mi = T_;
    for (int t = tid; t < T_; t += 256) {
        bool allowed = (t >= p) && ((T_ - 1 - t) >= mask_len) && (t < ml);
        float e = allowed ? energy[b * T_ + t] : NEGV;
        alignS[t] = e;
        if (e > mx || (e == mx && t < mi)) { mx = e; mi = t; }
    }
    red[tid] = mx; redi[tid] = mi;
    __syncthreads();
    for (int s = 128; s > 0; s >>= 1) {
        if (tid < s) {
            float o = red[tid + s]; int oi = redi[tid + s];
            if (o > red[tid] || (o == red[tid] && oi < redi[tid])) {
                red[tid] = o; redi[tid] = oi;
            }
        }
        __syncthreads();
    }
    const float gmax = red[0];
    const int   gargmax = redi[0];
    __syncthreads();

    // pass 2: exp + sum
    float psum = 0.f;
    for (int t = tid; t < T_; t += 256) {
        float ex = __expf(alignS[t] - gmax);
        alignS[t] = ex;
        psum += ex;
    }
    red[tid] = psum;
    __syncthreads();
    for (int s = 128; s > 0; s >>= 1) {
        if (tid < s) red[tid] += red[tid + s];
        __syncthreads();
    }
    const float inv = 1.f / red[0];
    __syncthreads();

    // pass 3: normalize, emit alignments + cumulative state
    for (int t = tid; t < T_; t += 256) {
        float a = alignS[t] * inv;
        alignS[t] = a;
        out_align[b * T_ + t] = a;
        out_next[b * T_ + t]  = a + prev_align[b * T_ + t];
    }
    __syncthreads();

    // pass 4: attention[b,d] = sum_t align[t]*values[b,t,d]  (coalesced over d)
    if (tid < D_) {
        float acc = 0.f;
        for (int t = 0; t < T_; ++t)
            acc = fmaf(alignS[t], values[(size_t)(b * T_ + t) * D_ + tid], acc);
        out_att[b * D_ + tid] = acc;
    }
    if (tid == 0) out_maxatt[b] = (float)gargmax;
}

// ---------------------------------------------------------------------------
extern "C" void kernel_launch(void* const* d_in, const int* in_sizes, int n_in,
                              void* d_out, int out_size, void* d_ws, size_t ws_size,
                              hipStream_t stream) {
    const float* query      = (const float*)d_in[0];
    const float* memory     = (const float*)d_in[1];
    const float* prev_align = (const float*)d_in[2];
    const int*   pma        = (const int*)d_in[3];
    const int*   memlen     = (const int*)d_in[4];
    const float* Wq         = (const float*)d_in[5];
    const float* Wm         = (const float*)d_in[6];
    const float* ck         = (const float*)d_in[7];
    const float* cb         = (const float*)d_in[8];
    const float* Wloc       = (const float*)d_in[9];
    const float* sv         = (const float*)d_in[10];
    const float* sb         = (const float*)d_in[11];

    // workspace layout (floats)
    float* values = (float*)d_ws;                         // 64*2000*128
    float* pB     = values + (size_t)B_ * T_ * D_;        // 512*128 packed
    float* pq     = pB + (size_t)DM_ * D_;                // 64*128
    float* energy = pq + (size_t)B_ * D_;                 // 64*2000

    // output layout (reference return order)
    float* out      = (float*)d_out;
    float* out_att  = out;                                // [64,128]
    float* out_algn = out_att + (size_t)B_ * D_;          // [64,2000]
    float* out_next = out_algn + (size_t)B_ * T_;         // [64,2000]
    float* out_marg = out_next + (size_t)B_ * T_;         // [64]

    pack_wm<<<(128 * 8 * 32 + 255) / 256, 256, 0, stream>>>(Wm, pB);
    values_gemm<<<(B_ * T_ / 16 + 7) / 8, 256, 0, stream>>>(memory, pB, values);
    pq_kernel<<<B_, 128, 0, stream>>>(query, Wq, pq);
    energy_kernel<<<B_ * (T_ / 16), 128, 0, stream>>>(prev_align, ck, cb, Wloc,
                                                      sv, sb, pq, values, energy);
    softmax_attn<<<B_, 256, 0, stream>>>(energy, prev_align, pma, memlen, values,
                                         out_att, out_algn, out_next, out_marg);
}